// CAM_Module_41094247088141
// MI455X (gfx1250) — compile-verified
//
#include <hip/hip_runtime.h>

typedef float v2f __attribute__((ext_vector_type(2)));
typedef float v8f __attribute__((ext_vector_type(8)));

#define B_DIM 32
#define C_DIM 512
#define N_DIM 4096

// ---------------------------------------------------------------------------
// gfx1250 async global->LDS copy (ASYNCcnt-tracked, no dest VGPRs).
// LDS byte offset = low 32 bits of the generic pointer (local->flat cast is
// {aperture_hi, lds_offset}).
// ---------------------------------------------------------------------------
__device__ __forceinline__ void async_ld_b128(void* lds, const void* gaddr) {
    asm volatile("global_load_async_to_lds_b128 %0, %1, off"
                 :
                 : "v"((unsigned int)(unsigned long long)lds), "v"(gaddr)
                 : "memory");
}
__device__ __forceinline__ void wait_asynccnt0() {
    asm volatile("s_wait_asynccnt 0x0" ::: "memory");
}

// ---------------------------------------------------------------------------
// Kernel 1: energy[b] = x[b] @ x[b]^T   (per-batch SYRK, fp32 WMMA 16x16x4)
// 128 threads (4 waves, 2x2), 64x64 tile, BK=16, double-buffered async LDS.
// Upper-triangular block tiles only; mirrored on store.
// ---------------------------------------------------------------------------
__global__ __launch_bounds__(128)
void cam_energy_kernel(const float* __restrict__ x, float* __restrict__ energy) {
    __shared__ float As[2][64 * 20];   // 64 rows x 16 K, stride 20 dw (80B, 16B-aligned)
    __shared__ float Bs[2][64 * 20];

    if (blockIdx.y < blockIdx.x) return;   // symmetry: bx <= by only

    const int b = blockIdx.z;
    const float* xb = x + (size_t)b * C_DIM * N_DIM;
    float* eb = energy + (size_t)b * C_DIM * C_DIM;

    const int tid  = threadIdx.x;
    const int lane = tid & 31;
    const int wave = tid >> 5;
    const int wm   = wave >> 1;
    const int wn   = wave & 1;
    const int rowBase = blockIdx.x * 64;
    const int colBase = blockIdx.y * 64;

    v8f acc[2][2];
#pragma unroll
    for (int i = 0; i < 2; ++i)
#pragma unroll
        for (int j = 0; j < 2; ++j)
#pragma unroll
            for (int e = 0; e < 8; ++e) acc[i][j][e] = 0.0f;

    const int half = lane >> 4;
    const int lm   = lane & 15;

    auto load_tiles = [&](int k0, int p) {
#pragma unroll
        for (int it = 0; it < 2; ++it) {
            int idx = tid + it * 128;          // 0..255 over 64 rows x 4 float4
            int r   = idx >> 2;
            int c4  = (idx & 3) << 2;
            async_ld_b128(&As[p][r * 20 + c4],
                          xb + (size_t)(rowBase + r) * N_DIM + k0 + c4);
            async_ld_b128(&Bs[p][r * 20 + c4],
                          xb + (size_t)(colBase + r) * N_DIM + k0 + c4);
        }
    };

    load_tiles(0, 0);
    wait_asynccnt0();
    __syncthreads();

    int p = 0;
    for (int k0 = 0; k0 < N_DIM; k0 += 16) {
        if (k0 + 16 < N_DIM) load_tiles(k0 + 16, p ^ 1);   // prefetch next tile

#pragma unroll
        for (int kk = 0; kk < 4; ++kk) {
            const int kidx = kk * 4 + half * 2;
            v2f a0 = *(const v2f*)(&As[p][(wm * 32 +      lm) * 20 + kidx]);
            v2f a1 = *(const v2f*)(&As[p][(wm * 32 + 16 + lm) * 20 + kidx]);
            v2f b0 = *(const v2f*)(&Bs[p][(wn * 32 +      lm) * 20 + kidx]);
            v2f b1 = *(const v2f*)(&Bs[p][(wn * 32 + 16 + lm) * 20 + kidx]);
            acc[0][0] = __builtin_amdgcn_wmma_f32_16x16x4_f32(false, a0, false, b0,
                        (short)0, acc[0][0], false, false);
            acc[0][1] = __builtin_amdgcn_wmma_f32_16x16x4_f32(false, a0, false, b1,
                        (short)0, acc[0][1], false, false);
            acc[1][0] = __builtin_amdgcn_wmma_f32_16x16x4_f32(false, a1, false, b0,
                        (short)0, acc[1][0], false, false);
            acc[1][1] = __builtin_amdgcn_wmma_f32_16x16x4_f32(false, a1, false, b1,
                        (short)0, acc[1][1], false, false);
        }

        wait_asynccnt0();   // next-tile async copies landed
        __syncthreads();    // everyone done reading cur / writing nxt
        p ^= 1;
    }

#pragma unroll
    for (int mi = 0; mi < 2; ++mi)
#pragma unroll
        for (int ni = 0; ni < 2; ++ni)
#pragma unroll
            for (int v = 0; v < 8; ++v) {
                int row = rowBase + wm * 32 + mi * 16 + v + half * 8;
                int col = colBase + wn * 32 + ni * 16 + lm;
                float val = acc[mi][ni][v];
                eb[(size_t)row * C_DIM + col] = val;
                eb[(size_t)col * C_DIM + row] = val;   // symmetric mirror
            }
}

// ---------------------------------------------------------------------------
// Kernel 2: attention = softmax(rowmax - energy) == softmax(-energy)
// One wave32 per 512-element row; stabilized with per-row min(energy).
// ---------------------------------------------------------------------------
__global__ __launch_bounds__(256)
void cam_softmax_kernel(float* __restrict__ energy) {
    const int row  = blockIdx.x * 8 + (threadIdx.x >> 5);
    const int lane = threadIdx.x & 31;
    float* e = energy + (size_t)row * C_DIM;

    float vals[16];
    float mn = 3.4e38f;
#pragma unroll
    for (int i = 0; i < 16; ++i) {
        vals[i] = e[lane + i * 32];
        mn = fminf(mn, vals[i]);
    }
#pragma unroll
    for (int off = 16; off; off >>= 1) mn = fminf(mn, __shfl_xor(mn, off, 32));

    float s = 0.0f;
#pragma unroll
    for (int i = 0; i < 16; ++i) {
        vals[i] = __expf(mn - vals[i]);   // exponent <= 0: stable
        s += vals[i];
    }
#pragma unroll
    for (int off = 16; off; off >>= 1) s += __shfl_xor(s, off, 32);

    const float inv = 1.0f / s;
#pragma unroll
    for (int i = 0; i < 16; ++i) e[lane + i * 32] = vals[i] * inv;
}

// ---------------------------------------------------------------------------
// Kernel 3: out[b] = gamma * (att[b] @ x[b]) + x[b]
// M=C=512, N=4096, K=C=512; same WMMA tiling, double-buffered async LDS,
// fused epilogue.
// ---------------------------------------------------------------------------
__global__ __launch_bounds__(128)
void cam_out_kernel(const float* __restrict__ att, const float* __restrict__ x,
                    const float* __restrict__ gamma, float* __restrict__ out) {
    __shared__ float As[2][64 * 20];   // att tile: 64 x 16, stride 20 dw
    __shared__ float Bs[2][16 * 68];   // x tile:   16 x 64, stride 68 dw (272B, 16B-aligned)

    const int b = blockIdx.z;
    const float* ab = att + (size_t)b * C_DIM * C_DIM;
    const float* xb = x   + (size_t)b * C_DIM * N_DIM;
    float*       ob = out + (size_t)b * C_DIM * N_DIM;

    const int tid  = threadIdx.x;
    const int lane = tid & 31;
    const int wave = tid >> 5;
    const int wm   = wave >> 1;
    const int wn   = wave & 1;
    const int rowBase = blockIdx.x * 64;   // over C (M)
    const int colBase = blockIdx.y * 64;   // over N

    v8f acc[2][2];
#pragma unroll
    for (int i = 0; i < 2; ++i)
#pragma unroll
        for (int j = 0; j < 2; ++j)
#pragma unroll
            for (int e = 0; e < 8; ++e) acc[i][j][e] = 0.0f;

    const int half = lane >> 4;
    const int lm   = lane & 15;

    auto load_tiles = [&](int k0, int p) {
#pragma unroll
        for (int it = 0; it < 2; ++it) {            // A: 64 rows x 4 float4
            int idx = tid + it * 128;
            int r   = idx >> 2;
            int c4  = (idx & 3) << 2;
            async_ld_b128(&As[p][r * 20 + c4],
                          ab + (size_t)(rowBase + r) * C_DIM + k0 + c4);
        }
#pragma unroll
        for (int it = 0; it < 2; ++it) {            // B: 16 rows x 16 float4
            int idx = tid + it * 128;
            int r   = idx >> 4;
            int c4  = (idx & 15) << 2;
            async_ld_b128(&Bs[p][r * 68 + c4],
                          xb + (size_t)(k0 + r) * N_DIM + colBase + c4);
        }
    };

    load_tiles(0, 0);
    wait_asynccnt0();
    __syncthreads();

    int p = 0;
    for (int k0 = 0; k0 < C_DIM; k0 += 16) {
        if (k0 + 16 < C_DIM) load_tiles(k0 + 16, p ^ 1);

#pragma unroll
        for (int kk = 0; kk < 4; ++kk) {
            const int kidx = kk * 4 + half * 2;
            v2f a0 = *(const v2f*)(&As[p][(wm * 32 +      lm) * 20 + kidx]);
            v2f a1 = *(const v2f*)(&As[p][(wm * 32 + 16 + lm) * 20 + kidx]);
            const int col0 = wn * 32 + lm;
            v2f b0 = { Bs[p][(kidx    ) * 68 + col0],
                       Bs[p][(kidx + 1) * 68 + col0] };
            v2f b1 = { Bs[p][(kidx    ) * 68 + col0 + 16],
                       Bs[p][(kidx + 1) * 68 + col0 + 16] };
            acc[0][0] = __builtin_amdgcn_wmma_f32_16x16x4_f32(false, a0, false, b0,
                        (short)0, acc[0][0], false, false);
            acc[0][1] = __builtin_amdgcn_wmma_f32_16x16x4_f32(false, a0, false, b1,
                        (short)0, acc[0][1], false, false);
            acc[1][0] = __builtin_amdgcn_wmma_f32_16x16x4_f32(false, a1, false, b0,
                        (short)0, acc[1][0], false, false);
            acc[1][1] = __builtin_amdgcn_wmma_f32_16x16x4_f32(false, a1, false, b1,
                        (short)0, acc[1][1], false, false);
        }

        wait_asynccnt0();
        __syncthreads();
        p ^= 1;
    }

    const float g = gamma[0];
#pragma unroll
    for (int mi = 0; mi < 2; ++mi)
#pragma unroll
        for (int ni = 0; ni < 2; ++ni)
#pragma unroll
            for (int v = 0; v < 8; ++v) {
                int row = rowBase + wm * 32 + mi * 16 + v + half * 8;
                int col = colBase + wn * 32 + ni * 16 + lm;
                size_t off = (size_t)row * N_DIM + col;
                ob[off] = g * acc[mi][ni][v] + xb[off];
            }
}

// ---------------------------------------------------------------------------
extern "C" void kernel_launch(void* const* d_in, const int* in_sizes, int n_in,
                              void* d_out, int out_size, void* d_ws, size_t ws_size,
                              hipStream_t stream) {
    const float* x     = (const float*)d_in[0];   // [B, C, N] fp32
    const float* gamma = (const float*)d_in[1];   // [1] fp32
    float*       out   = (float*)d_out;           // [B, C, N] fp32
    float*       energy = (float*)d_ws;           // [B, C, C] fp32 = 32 MB scratch

    dim3 g1(C_DIM / 64, C_DIM / 64, B_DIM);       // (8, 8, 32)
    cam_energy_kernel<<<g1, 128, 0, stream>>>(x, energy);

    cam_softmax_kernel<<<(B_DIM * C_DIM) / 8, 256, 0, stream>>>(energy);

    dim3 g3(C_DIM / 64, N_DIM / 64, B_DIM);       // (8, 64, 32)
    cam_out_kernel<<<g3, 128, 0, stream>>>(energy, x, gamma, out);
}